// TopkGCN_29334626631946
// MI455X (gfx1250) — compile-verified
//
#include <hip/hip_runtime.h>
#include <math.h>

// ---------------- constants from the reference ----------------
#define BG    64            // graphs
#define NPG0  1000          // nodes per graph at layer 0
#define HD    128           // hidden dim
#define IND   64            // input dim
#define EDG   (64*1000*16)  // edges = 1,024,000
#define CNUM  10
#define EPSBN 1e-5f
#define NEGINF (-3.402823466e38f)

typedef __attribute__((ext_vector_type(2))) float v2f;
typedef __attribute__((ext_vector_type(8))) float v8f;

// ---------------- small utility kernels ----------------
__global__ void zero_f(float* __restrict__ p, int n) {
    int i = blockIdx.x * blockDim.x + threadIdx.x;
    if (i < n) p[i] = 0.f;
}
__global__ void fill_f(float* __restrict__ p, float v, int n) {
    int i = blockIdx.x * blockDim.x + threadIdx.x;
    if (i < n) p[i] = v;
}
__global__ void rsqrt_k(float* __restrict__ p, int n) {
    int i = blockIdx.x * blockDim.x + threadIdx.x;
    if (i < n) p[i] = rsqrtf(p[i]);
}

// ---------------- fp32 WMMA GEMM: Out[N,HO] = A[N,K] @ W[K,HO] (+bias, relu) ----
// Block = 256 threads = 8 waves. Block computes a 16-row strip x 128 cols
// (grid.y selects 128-col slab). Each wave owns a 16x16 tile and iterates K
// in steps of 4 with V_WMMA_F32_16X16X4_F32 (fp32-exact vs reference).
__global__ __launch_bounds__(256) void gemm_wmma(
    const float* __restrict__ A, const float* __restrict__ W,
    float* __restrict__ Out, int N, int K, int HO,
    const float* __restrict__ bias, int relu)
{
    __shared__ float ldsA[16 * 256];   // 16 rows x K (K <= 256) = 16KB max
    const int tid  = threadIdx.x;
    const int row0 = blockIdx.x * 16;

    // Stage the 16xK A-tile into LDS (zero-pad rows beyond N).
    for (int t = tid; t < 16 * K; t += 256) {
        int r = t / K, c = t - r * K;
        int gr = row0 + r;
        ldsA[t] = (gr < N) ? A[(size_t)gr * K + c] : 0.f;
    }
    __syncthreads();

    const int wave = tid >> 5;
    const int lane = tid & 31;
    const int m    = lane & 15;          // row within tile (A) / col (B,D)
    const int hi   = lane >> 4;          // half-wave selects K pair / row group
    const int colg = blockIdx.y * 128 + wave * 16 + m;

    v8f acc = {};
    for (int k0 = 0; k0 < K; k0 += 4) {
        const int kk = k0 + (hi << 1);   // lanes 0-15: k0,k0+1 ; lanes 16-31: k0+2,k0+3
        v2f a, b;
        a.x = ldsA[m * K + kk];
        a.y = ldsA[m * K + kk + 1];
        b.x = W[(size_t)kk * HO + colg];
        b.y = W[(size_t)(kk + 1) * HO + colg];
        acc = __builtin_amdgcn_wmma_f32_16x16x4_f32(
            /*neg_a=*/false, a, /*neg_b=*/false, b,
            /*c_mod=*/(short)0, acc, /*reuse_a=*/false, /*reuse_b=*/false);
    }

    const int rbase = row0 + (hi << 3);  // lanes 0-15 -> M=v, lanes 16-31 -> M=v+8
    for (int v = 0; v < 8; ++v) {
        int r = rbase + v;
        if (r < N) {
            float o = acc[v];
            if (bias) o += bias[colg];
            if (relu) o = fmaxf(o, 0.f);
            Out[(size_t)r * HO + colg] = o;
        }
    }
}

// ---------------- degree accumulation ----------------
__global__ void edge_deg(const int* __restrict__ dst, const float* __restrict__ we,
                         float* __restrict__ deg, int E) {
    int e = blockIdx.x * blockDim.x + threadIdx.x;
    if (e < E) atomicAdd(&deg[dst[e]], we[e]);
}

// ---------------- edge scatter: agg[dst] += coef * xw[src] ----------------
// One wave32 per edge, 4 features per lane. Memory-bound: ~1 GB/layer.
__global__ __launch_bounds__(256) void scatter_k(
    const float* __restrict__ xw, const int* __restrict__ src,
    const int* __restrict__ dst, const float* __restrict__ we,
    const float* __restrict__ dinv, float* __restrict__ agg, int E)
{
    int e    = (blockIdx.x * blockDim.x + threadIdx.x) >> 5;
    int lane = threadIdx.x & 31;
    if (e >= E) return;
    float w = we[e];
    if (w == 0.f) return;                 // dead / filtered edge
    int s = src[e], d = dst[e];
    float coef = dinv[s] * dinv[d] * w;
    #pragma unroll
    for (int c = 0; c < 4; ++c) {
        int f = lane + 32 * c;
        atomicAdd(&agg[(size_t)d * HD + f], coef * xw[(size_t)s * HD + f]);
    }
}

// ---------------- self-loop + bias + BN + ReLU (in place on agg) -------------
__global__ void combine_k(float* __restrict__ agg, const float* __restrict__ xw,
                          const float* __restrict__ dinv,
                          const float* __restrict__ b, const float* __restrict__ g,
                          const float* __restrict__ bt, const float* __restrict__ mu,
                          const float* __restrict__ var, int N)
{
    int i = blockIdx.x * blockDim.x + threadIdx.x;
    if (i >= N * HD) return;
    int n = i >> 7, f = i & 127;
    float di = dinv[n];
    float v = agg[i] + di * di * xw[i] + b[f];
    v = (v - mu[f]) * (g[f] * rsqrtf(var[f] + EPSBN)) + bt[f];
    agg[i] = fmaxf(v, 0.f);
}

// ---------------- pool weight inverse norm (single wave) ----------------
__global__ void prep_pw(const float* __restrict__ pw, float* __restrict__ pwinv) {
    int lane = threadIdx.x;
    float s = 0.f;
    #pragma unroll
    for (int c = 0; c < 4; ++c) { float v = pw[lane + 32 * c]; s += v * v; }
    #pragma unroll
    for (int m = 16; m >= 1; m >>= 1) s += __shfl_xor(s, m, 32);
    if (lane == 0) pwinv[0] = rsqrtf(s);
}

// ---------------- node score: tanh(dot(hc, pw)/||pw||), one wave per node ----
__global__ __launch_bounds__(256) void score_k(
    const float* __restrict__ hc, const float* __restrict__ pw,
    const float* __restrict__ pwinv, float* __restrict__ score, int N)
{
    int node = (blockIdx.x * blockDim.x + threadIdx.x) >> 5;
    int lane = threadIdx.x & 31;
    if (node >= N) return;
    float dot = 0.f;
    #pragma unroll
    for (int c = 0; c < 4; ++c) {
        int f = lane + 32 * c;
        dot += hc[(size_t)node * HD + f] * pw[f];
    }
    #pragma unroll
    for (int m = 16; m >= 1; m >>= 1) dot += __shfl_xor(dot, m, 32);
    if (lane == 0) score[node] = tanhf(dot * pwinv[0]);
}

// ---------------- per-graph top-k via 1024-wide LDS bitonic sort -------------
// Descending by score, tie-break ascending index (mirrors jax top_k stability).
__global__ __launch_bounds__(512) void topk_k(
    const float* __restrict__ score, int n, int k,
    int* __restrict__ perm, float* __restrict__ vals, int* __restrict__ inv)
{
    __shared__ float s[1024];
    __shared__ int   id[1024];
    const int g = blockIdx.x, tid = threadIdx.x;
    for (int t = tid; t < 1024; t += 512) {
        s[t]  = (t < n) ? score[g * n + t] : NEGINF;
        id[t] = t;
    }
    for (int t = tid; t < n; t += 512) inv[g * n + t] = -1;
    __syncthreads();

    for (int ksz = 2; ksz <= 1024; ksz <<= 1) {
        for (int j = ksz >> 1; j > 0; j >>= 1) {
            for (int t = tid; t < 1024; t += 512) {
                int l = t ^ j;
                if (l > t) {
                    bool desc  = ((t & ksz) == 0);
                    bool after = (s[t] < s[l]) || ((s[t] == s[l]) && (id[t] > id[l]));
                    if (desc ? after : !after) {
                        float ts = s[t]; s[t] = s[l]; s[l] = ts;
                        int   ti = id[t]; id[t] = id[l]; id[l] = ti;
                    }
                }
            }
            __syncthreads();
        }
    }
    for (int t = tid; t < k; t += 512) {
        perm[g * k + t] = g * n + id[t];
        vals[g * k + t] = s[t];
        inv[g * n + id[t]] = g * k + t;
    }
}

// ---------------- gather + scale + mean/max pooling into flats ---------------
__global__ __launch_bounds__(128) void pool_k(
    const float* __restrict__ hc, const int* __restrict__ perm,
    const float* __restrict__ vals, float* __restrict__ hnew,
    float* __restrict__ flats, int k)
{
    const int g = blockIdx.x, f = threadIdx.x;
    float sum = 0.f, mx = NEGINF;
    for (int j = 0; j < k; ++j) {
        int p = perm[g * k + j];
        float v = hc[(size_t)p * HD + f] * vals[g * k + j];
        hnew[(size_t)(g * k + j) * HD + f] = v;
        sum += v;
        mx = fmaxf(mx, v);
    }
    flats[g * 256 + f]       += sum / (float)k;
    flats[g * 256 + 128 + f] += mx;
}

// ---------------- edge remap through inv ----------------
__global__ void remap_k(const int* __restrict__ srcI, const int* __restrict__ dstI,
                        const float* __restrict__ weI, const int* __restrict__ inv,
                        int* __restrict__ srcO, int* __restrict__ dstO,
                        float* __restrict__ weO, int E)
{
    int e = blockIdx.x * blockDim.x + threadIdx.x;
    if (e >= E) return;
    int s = inv[srcI[e]], d = inv[dstI[e]];
    bool ok = (s >= 0) && (d >= 0);
    srcO[e] = ok ? s : 0;
    dstO[e] = ok ? d : 0;
    weO[e]  = ok ? weI[e] : 0.f;
}

// ---------------- tiny final dense: out[64,10] = hd @ d2_w + d2_b ------------
__global__ void head2_k(const float* __restrict__ hd, const float* __restrict__ w,
                        const float* __restrict__ b, float* __restrict__ out)
{
    int i = blockIdx.x * blockDim.x + threadIdx.x;
    if (i >= BG * CNUM) return;
    int g = i / CNUM, c = i - g * CNUM;
    float s = b[c];
    for (int kk = 0; kk < 512; ++kk) s += hd[g * 512 + kk] * w[kk * CNUM + c];
    out[i] = s;
}

// ---------------- host-side orchestration ----------------
extern "C" void kernel_launch(void* const* d_in, const int* in_sizes, int n_in,
                              void* d_out, int out_size, void* d_ws, size_t ws_size,
                              hipStream_t stream) {
    const float* x       = (const float*)d_in[0];
    const int*   ei      = (const int*)  d_in[1];
    const float* ew      = (const float*)d_in[3];
    const float* conv1_w = (const float*)d_in[4];
    const float* conv_w  = (const float*)d_in[5];
    const float* conv_b  = (const float*)d_in[6];
    const float* bn_g    = (const float*)d_in[7];
    const float* bn_b    = (const float*)d_in[8];
    const float* bn_m    = (const float*)d_in[9];
    const float* bn_v    = (const float*)d_in[10];
    const float* pool_w  = (const float*)d_in[11];
    const float* d1_w    = (const float*)d_in[12];
    const float* d1_b    = (const float*)d_in[13];
    const float* d2_w    = (const float*)d_in[14];
    const float* d2_b    = (const float*)d_in[15];
    float* out = (float*)d_out;

    // workspace carve-up
    char* ws = (char*)d_ws;
    size_t off = 0;
    auto allocF = [&](size_t n) -> float* {
        float* p = (float*)(ws + off); off += ((n * 4 + 255) / 256) * 256; return p; };
    auto allocI = [&](size_t n) -> int* {
        int* p = (int*)(ws + off); off += ((n * 4 + 255) / 256) * 256; return p; };

    float* xw    = allocF((size_t)BG * NPG0 * HD);   // 32.8 MB
    float* agg   = allocF((size_t)BG * NPG0 * HD);   // 32.8 MB (becomes hc)
    float* hA    = allocF((size_t)BG * 800 * HD);    // 26.2 MB
    float* hB    = allocF((size_t)BG * 800 * HD);    // 26.2 MB
    float* deg   = allocF(BG * NPG0);                // deg -> dinv in place
    float* score = allocF(BG * NPG0);
    float* vals  = allocF(BG * 800);
    float* flats = allocF(BG * 256);
    float* hd    = allocF(BG * 512);
    float* pwinv = allocF(64);
    int*   perm  = allocI(BG * 800);
    int*   inv   = allocI(BG * NPG0);
    int*   srcP[2] = { allocI(EDG), allocI(EDG) };
    int*   dstP[2] = { allocI(EDG), allocI(EDG) };
    float* weP [2] = { allocF(EDG), allocF(EDG) };
    float* hbuf[2] = { hA, hB };
    (void)ws_size; (void)in_sizes; (void)n_in; (void)out_size;

    const int ncur[7] = {1000, 800, 640, 512, 410, 328, 263}; // k = ceil(0.8*n)

    zero_f<<<(BG * 256 + 255) / 256, 256, 0, stream>>>(flats, BG * 256);

    const float* hcur = x;
    const int*   srcC = ei;
    const int*   dstC = ei + EDG;
    const float* weC  = ew;

    for (int i = 0; i < 6; ++i) {
        const int n = ncur[i], kk = ncur[i + 1];
        const int N = BG * n;
        const int K = (i == 0) ? IND : HD;
        const float* W = (i == 0) ? conv1_w : conv_w + (size_t)(i - 1) * HD * HD;

        // xw = h @ W  (WMMA fp32)
        gemm_wmma<<<dim3((N + 15) / 16, 1), 256, 0, stream>>>(
            hcur, W, xw, N, K, HD, nullptr, 0);

        // deg -> dinv
        fill_f<<<(N + 255) / 256, 256, 0, stream>>>(deg, 1.0f, N);
        edge_deg<<<(EDG + 255) / 256, 256, 0, stream>>>(dstC, weC, deg, EDG);
        rsqrt_k<<<(N + 255) / 256, 256, 0, stream>>>(deg, N);

        // edge aggregation
        zero_f<<<(N * HD + 255) / 256, 256, 0, stream>>>(agg, N * HD);
        scatter_k<<<EDG / 8, 256, 0, stream>>>(xw, srcC, dstC, weC, deg, agg, EDG);

        // self loop + bias + BN + ReLU (agg becomes hc)
        combine_k<<<(N * HD + 255) / 256, 256, 0, stream>>>(
            agg, xw, deg, conv_b + i * HD, bn_g + i * HD, bn_b + i * HD,
            bn_m + i * HD, bn_v + i * HD, N);

        // scores + per-graph top-k
        prep_pw<<<1, 32, 0, stream>>>(pool_w + i * HD, pwinv);
        score_k<<<(N + 7) / 8, 256, 0, stream>>>(agg, pool_w + i * HD, pwinv, score, N);
        topk_k<<<BG, 512, 0, stream>>>(score, n, kk, perm, vals, inv);

        // pooled gather into next-layer h, accumulate flats
        float* hnext = hbuf[i & 1];
        pool_k<<<BG, 128, 0, stream>>>(agg, perm, vals, hnext, flats, kk);

        // remap edges for next layer
        if (i < 5) {
            int* sN = srcP[i & 1]; int* dN = dstP[i & 1]; float* wN = weP[i & 1];
            remap_k<<<(EDG + 255) / 256, 256, 0, stream>>>(
                srcC, dstC, weC, inv, sN, dN, wN, EDG);
            srcC = sN; dstC = dN; weC = wN;
        }
        hcur = hnext;
    }

    // MLP head: hd = relu(flats @ d1_w + d1_b) via WMMA, then tiny dense to C=10
    gemm_wmma<<<dim3((BG + 15) / 16, 4), 256, 0, stream>>>(
        flats, d1_w, hd, BG, 256, 512, d1_b, 1);
    head2_k<<<(BG * CNUM + 255) / 256, 256, 0, stream>>>(hd, d2_w, d2_b, out);
}